// NonMaxSuppression_83958020702833
// MI455X (gfx1250) — compile-verified
//
#include <hip/hip_runtime.h>
#include <hip/hip_bf16.h>

typedef unsigned int u32;
typedef unsigned long long u64;

typedef float v2f  __attribute__((ext_vector_type(2)));
typedef float v8f  __attribute__((ext_vector_type(8)));
typedef u32   u32x4 __attribute__((ext_vector_type(4)));
typedef int   i32x4 __attribute__((ext_vector_type(4)));
typedef int   i32x8 __attribute__((ext_vector_type(8)));

#define NBOX 6144
#define NW   192   // 6144 / 32 mask words per row

// ---------------------------------------------------------------------------
// Kernel 1: stable descending rank by confidence (ties broken by index, which
// matches jnp.argsort(-conf) stable semantics). Confidences (stride-5 inside
// preds) are pulled into LDS via the Tensor Data Mover (2D descriptor:
// dim0=1 elem, stride0=5 elems, 6144 rows). A plain-load pass writes the same
// values for belt-and-braces correctness. The rank scan reads LDS 16B at a
// time (ds_load_b128) via a float4-typed buffer.
// ---------------------------------------------------------------------------
__global__ __launch_bounds__(256) void rank_kernel(const float* __restrict__ preds,
                                                   float4* __restrict__ sortedBoxes,
                                                   int* __restrict__ sortedIdx) {
  __shared__ float4 conf4[NBOX / 4];   // 24 KB, 16B-aligned
  float* conf = (float*)conf4;

  if (threadIdx.x < 32) {        // wave 0 issues one TDM load for the block
    u64 ga  = (u64)(uintptr_t)preds + 16ull;          // &preds[0][4], byte addr
    u32 lds = (u32)(uintptr_t)conf;                   // low 32 bits = LDS addr
    u32x4 g0;
    g0[0] = 1u;                                       // count = 1 valid D#
    g0[1] = lds;                                      // lds_addr
    g0[2] = (u32)(ga & 0xFFFFFFFFull);                // global_addr[31:0]
    g0[3] = (u32)((ga >> 32) & 0x01FFFFFFull) | (2u << 30); // addr[56:32] | type=2
    i32x8 g1;
    g1[0] = (int)(2u << 16);                          // data_size = 2 (4 bytes)
    g1[1] = (int)((1u & 0xFFFFu) << 16);              // tensor_dim0[15:0] = 1
    g1[2] = (int)((NBOX & 0xFFFFu) << 16);            // dim0 hi=0 | tensor_dim1 lo
    g1[3] = (int)((NBOX >> 16) | (1u << 16));         // dim1 hi | tile_dim0 = 1
    g1[4] = (int)(NBOX);                              // tile_dim1 = 6144, tile_dim2 = 0
    g1[5] = 5;                                        // tensor_dim0_stride = 5 elems
    g1[6] = 0;
    g1[7] = 0;
    i32x4 gz = {0, 0, 0, 0};
#if __clang_major__ >= 23
    i32x8 gz8 = {0, 0, 0, 0, 0, 0, 0, 0};
    __builtin_amdgcn_tensor_load_to_lds(g0, g1, gz, gz, gz8, 0);
#else
    __builtin_amdgcn_tensor_load_to_lds(g0, g1, gz, gz, 0);
#endif
    __builtin_amdgcn_s_wait_tensorcnt(0);
  }
  __syncthreads();

  // Redundant plain fill with identical values (guarantees LDS contents).
  for (int j = threadIdx.x; j < NBOX; j += 256) conf[j] = preds[j * 5 + 4];
  __syncthreads();

  const int i = blockIdx.x * 256 + threadIdx.x;
  const float ci = conf[i];
  int r = 0;
#pragma unroll 4
  for (int j4 = 0; j4 < NBOX / 4; ++j4) {
    float4 c = conf4[j4];
    int j = j4 * 4;
    r += (c.x > ci) || (c.x == ci && (j + 0) < i);
    r += (c.y > ci) || (c.y == ci && (j + 1) < i);
    r += (c.z > ci) || (c.z == ci && (j + 2) < i);
    r += (c.w > ci) || (c.w == ci && (j + 3) < i);
  }
  const float* p = preds + (size_t)i * 5;
  sortedBoxes[r] = make_float4(p[0], p[1], p[2], p[3]);
  sortedIdx[r] = i;
}

// ---------------------------------------------------------------------------
// Kernel 2: suppression bit-mask. Each wave owns one 16-row x 32-col region:
// two V_WMMA_F32_16X16X4_F32 ops compute the area[m]+area[n] outer-sum
// (A row = [area_m, 1, 0, 0], B col = [1, area_n, 0, 0]); intersections are
// computed on the WMMA C/D lane layout (VGPR v, lane L -> row v+(L<16?0:8),
// col L&15) and ballot_w32 packs the 16+16 bits per row into one mask word.
// All 8 accumulator slots use NAMED registers (row boxes preloaded, WMMA
// elements extracted to scalars, ballots in named u32s) so the compiler
// cannot re-roll the slots into a loop that re-executes the WMMAs.
// ---------------------------------------------------------------------------
__global__ __launch_bounds__(256) void mask_kernel(const float4* __restrict__ sortedBoxes,
                                                   const float* __restrict__ threshp,
                                                   u32* __restrict__ mask) {
  const int rowTile = blockIdx.y;          // 0..383
  const int m0 = rowTile * 16;
  const int wave = threadIdx.x >> 5;
  const int lane = threadIdx.x & 31;
  const int colWord = blockIdx.x * 8 + wave;   // 0..191

  __shared__ float4 rowBox[16];
  __shared__ float  rowArea[16];
  if (threadIdx.x < 16) {
    float4 b = sortedBoxes[m0 + threadIdx.x];
    rowBox[threadIdx.x] = b;
    rowArea[threadIdx.x] = (b.z - b.x) * (b.w - b.y);
  }
  __syncthreads();

  // Words strictly below the diagonal are never read by the reduction.
  if (colWord * 32 + 32 <= m0) return;

  const float thresh = *threshp;
  const int nlo = lane & 15;
  const bool lo = lane < 16;

  float4 bA = sortedBoxes[colWord * 32 + nlo];
  float4 bB = sortedBoxes[colWord * 32 + 16 + nlo];
  float areaA = (bA.z - bA.x) * (bA.w - bA.y);
  float areaB = (bB.z - bB.x) * (bB.w - bB.y);

  // Preload the 8 row boxes this half-lane needs into named registers.
  const int base = lo ? 0 : 8;
  float4 rb0 = rowBox[base + 0], rb1 = rowBox[base + 1];
  float4 rb2 = rowBox[base + 2], rb3 = rowBox[base + 3];
  float4 rb4 = rowBox[base + 4], rb5 = rowBox[base + 5];
  float4 rb6 = rowBox[base + 6], rb7 = rowBox[base + 7];

  v2f a, vbA, vbB;
  a[0]   = lo ? rowArea[nlo] : 0.0f;   // A[m,0] = area_m
  a[1]   = lo ? 1.0f : 0.0f;           // A[m,1] = 1
  vbA[0] = lo ? 1.0f : 0.0f;           // B[0,n] = 1
  vbA[1] = lo ? areaA : 0.0f;          // B[1,n] = area_n
  vbB[0] = lo ? 1.0f : 0.0f;
  vbB[1] = lo ? areaB : 0.0f;

  v8f zero = {};
  v8f sumA = __builtin_amdgcn_wmma_f32_16x16x4_f32(false, a, false, vbA,
                                                   (short)0, zero, false, false);
  v8f sumB = __builtin_amdgcn_wmma_f32_16x16x4_f32(false, a, false, vbB,
                                                   (short)0, zero, false, false);

  // Extract accumulator elements to named scalars (constant indices).
  float sa0 = sumA[0], sa1 = sumA[1], sa2 = sumA[2], sa3 = sumA[3];
  float sa4 = sumA[4], sa5 = sumA[5], sa6 = sumA[6], sa7 = sumA[7];
  float sb0 = sumB[0], sb1 = sumB[1], sb2 = sumB[2], sb3 = sumB[3];
  float sb4 = sumB[4], sb5 = sumB[5], sb6 = sumB[6], sb7 = sumB[7];

  u32 a0, a1, a2, a3, a4, a5, a6, a7;
  u32 b0, b1, b2, b3, b4, b5, b6, b7;

#define NMS_STEP(RB, SA, SB, BALA, BALB)                                       \
  do {                                                                         \
    float ix1 = fmaxf((RB).x, bA.x), iy1 = fmaxf((RB).y, bA.y);                \
    float ix2 = fminf((RB).z, bA.z), iy2 = fminf((RB).w, bA.w);                \
    float inter = fmaxf(ix2 - ix1, 0.0f) * fmaxf(iy2 - iy1, 0.0f);             \
    bool sA = inter > thresh * fmaxf((SA) - inter, 1e-9f);                     \
    BALA = __builtin_amdgcn_ballot_w32(sA);                                    \
    ix1 = fmaxf((RB).x, bB.x); iy1 = fmaxf((RB).y, bB.y);                      \
    ix2 = fminf((RB).z, bB.z); iy2 = fminf((RB).w, bB.w);                      \
    inter = fmaxf(ix2 - ix1, 0.0f) * fmaxf(iy2 - iy1, 0.0f);                   \
    bool sB = inter > thresh * fmaxf((SB) - inter, 1e-9f);                     \
    BALB = __builtin_amdgcn_ballot_w32(sB);                                    \
  } while (0)

  NMS_STEP(rb0, sa0, sb0, a0, b0);
  NMS_STEP(rb1, sa1, sb1, a1, b1);
  NMS_STEP(rb2, sa2, sb2, a2, b2);
  NMS_STEP(rb3, sa3, sb3, a3, b3);
  NMS_STEP(rb4, sa4, sb4, a4, b4);
  NMS_STEP(rb5, sa5, sb5, a5, b5);
  NMS_STEP(rb6, sa6, sb6, a6, b6);
  NMS_STEP(rb7, sa7, sb7, a7, b7);
#undef NMS_STEP

  if (lane == 0) {
    u32* rowp = mask + (size_t)m0 * NW + colWord;
    // ballot bits [15:0] = row v (cols 0-15), [31:16] = row v+8 (cols 0-15)
    rowp[0 * NW]  = (a0 & 0xFFFFu) | (b0 << 16);
    rowp[1 * NW]  = (a1 & 0xFFFFu) | (b1 << 16);
    rowp[2 * NW]  = (a2 & 0xFFFFu) | (b2 << 16);
    rowp[3 * NW]  = (a3 & 0xFFFFu) | (b3 << 16);
    rowp[4 * NW]  = (a4 & 0xFFFFu) | (b4 << 16);
    rowp[5 * NW]  = (a5 & 0xFFFFu) | (b5 << 16);
    rowp[6 * NW]  = (a6 & 0xFFFFu) | (b6 << 16);
    rowp[7 * NW]  = (a7 & 0xFFFFu) | (b7 << 16);
    rowp[8 * NW]  = (a0 >> 16) | (b0 & 0xFFFF0000u);
    rowp[9 * NW]  = (a1 >> 16) | (b1 & 0xFFFF0000u);
    rowp[10 * NW] = (a2 >> 16) | (b2 & 0xFFFF0000u);
    rowp[11 * NW] = (a3 >> 16) | (b3 & 0xFFFF0000u);
    rowp[12 * NW] = (a4 >> 16) | (b4 & 0xFFFF0000u);
    rowp[13 * NW] = (a5 >> 16) | (b5 & 0xFFFF0000u);
    rowp[14 * NW] = (a6 >> 16) | (b6 & 0xFFFF0000u);
    rowp[15 * NW] = (a7 >> 16) | (b7 & 0xFFFF0000u);
  }
}

// ---------------------------------------------------------------------------
// Kernel 3: greedy sequential reduction by a single wave32. The 192 removal
// words live in registers (6 per lane); the "row i alive" bit is broadcast
// with a shuffle each iteration; kept rows OR their (j>i)-masked mask row in.
// Then lane 0 applies the max_proposals cap via a running popcount.
// ---------------------------------------------------------------------------
__global__ void reduce_kernel(const u32* __restrict__ mask,
                              const int* __restrict__ maxPropP,
                              u32* __restrict__ keepWords) {
  const int lane = threadIdx.x;   // 0..31; lane owns words lane + s*32, s=0..5
  u32 rem[6] = {0, 0, 0, 0, 0, 0};

#pragma unroll
  for (int slot = 0; slot < 6; ++slot) {
    for (int i = slot << 10; i < ((slot + 1) << 10); ++i) {
      const int w = i >> 5;
      u32 rword = (u32)__shfl((int)rem[slot], w & 31, 32);
      if ((rword >> (i & 31)) & 1u) continue;     // row i already suppressed
      const u32* row = mask + (size_t)i * NW;
      u32 lowmask = (u32)((2ull << (i & 31)) - 1ull);  // bits j<=i within word w
#pragma unroll
      for (int s = 0; s < 6; ++s) {
        int wi = lane + (s << 5);
        if (wi > w)       rem[s] |= row[wi];
        else if (wi == w) rem[s] |= row[wi] & ~lowmask;
      }
    }
  }

  __shared__ u32 kw[NW];
#pragma unroll
  for (int s = 0; s < 6; ++s) kw[lane + (s << 5)] = ~rem[s];
  __syncthreads();

  if (lane == 0) {
    int cap = *maxPropP;
    if (cap > 0) {
      int run = 0;
      for (int w = 0; w < NW; ++w) {
        u32 word = kw[w];
        int c = __popc(word);
        if (run + c > cap) {
          int need = cap - run;
          u32 nw = 0;
          for (int b = 0; b < 32 && need > 0; ++b)
            if ((word >> b) & 1u) { nw |= 1u << b; --need; }
          word = nw;
        }
        run += __popc(word);
        kw[w] = word;
      }
    }
  }
  __syncthreads();
#pragma unroll
  for (int s = 0; s < 6; ++s) keepWords[lane + (s << 5)] = kw[lane + (s << 5)];
}

// ---------------------------------------------------------------------------
// Kernel 4: emit (kept_preds, keep_idx) in sorted order. thresh<0 passthrough
// returns preds unchanged (single-output path).
// ---------------------------------------------------------------------------
__global__ __launch_bounds__(256) void finalize_kernel(const float* __restrict__ preds,
                                                       const float* __restrict__ threshp,
                                                       const u32* __restrict__ keepWords,
                                                       const int* __restrict__ sortedIdx,
                                                       float* __restrict__ out,
                                                       int* __restrict__ outIdx) {
  const int s = blockIdx.x * 256 + threadIdx.x;
  if (*threshp < 0.0f) {
#pragma unroll
    for (int k = 0; k < 5; ++k) out[(size_t)s * 5 + k] = preds[(size_t)s * 5 + k];
    return;
  }
  const bool kept = (keepWords[s >> 5] >> (s & 31)) & 1u;
  const int orig = sortedIdx[s];
  const float* p = preds + (size_t)orig * 5;
  const float scale = kept ? 1.0f : 0.0f;
#pragma unroll
  for (int k = 0; k < 5; ++k) out[(size_t)s * 5 + k] = p[k] * scale;
  outIdx[s] = kept ? orig : -1;
}

// ---------------------------------------------------------------------------
extern "C" void kernel_launch(void* const* d_in, const int* in_sizes, int n_in,
                              void* d_out, int out_size, void* d_ws, size_t ws_size,
                              hipStream_t stream) {
  const float* preds   = (const float*)d_in[0];   // 6144 x 5 f32
  const float* thresh  = (const float*)d_in[1];   // scalar f32
  const int*   maxProp = (const int*)d_in[2];     // scalar i32

  char* ws = (char*)d_ws;
  float4* sortedBoxes = (float4*)(ws);                 //  98304 B
  int*    sortedIdx   = (int*)(ws + 98304);            //  24576 B
  u32*    keepWords   = (u32*)(ws + 122880);           //    768 B
  u32*    mask        = (u32*)(ws + 123904);           // 4718592 B (aligned 256)

  float* out    = (float*)d_out;                       // 6144 x 5 kept_preds
  int*   outIdx = (int*)d_out + (size_t)NBOX * 5;      // 6144 keep_idx

  rank_kernel<<<NBOX / 256, 256, 0, stream>>>(preds, sortedBoxes, sortedIdx);
  mask_kernel<<<dim3(NW / 8, NBOX / 16), 256, 0, stream>>>(sortedBoxes, thresh, mask);
  reduce_kernel<<<1, 32, 0, stream>>>(mask, maxProp, keepWords);
  finalize_kernel<<<NBOX / 256, 256, 0, stream>>>(preds, thresh, keepWords,
                                                  sortedIdx, out, outIdx);
}